// PrototypicalLoss_83966610637134
// MI455X (gfx1250) — compile-verified
//
#include <hip/hip_runtime.h>

typedef float v2f __attribute__((ext_vector_type(2)));
typedef float v8f __attribute__((ext_vector_type(8)));

#define DIM   256
#define NWAY  5
#define QTILE 16
#define PTILE (QTILE * NWAY)   // 80 prompts per tile

// One wave (32 threads) per block; each block handles 16 queries + their 80 prototypes.
__global__ __launch_bounds__(32) void proto_main_kernel(
    const float* __restrict__ qry,     // [n, 256]
    const float* __restrict__ prm,     // [n*5, 256]
    const float* __restrict__ lab,     // [n, 5] one-hot
    float* __restrict__ partial)       // [gridDim.x * 2] : (loss_sum, correct_sum)
{
    const int lane  = threadIdx.x;        // 0..31
    const int tile  = blockIdx.x;         // tile of 16 queries
    const int row   = lane & 15;          // matrix row this lane feeds
    const int khalf = (lane >> 4) << 1;   // lanes 0-15 -> K+0/K+1, lanes 16-31 -> K+2/K+3

    // A-fragment source: query row `row` of this tile
    const float* qbase = qry + (size_t)(tile * QTILE + row) * DIM + khalf;
    // B-fragment sources: 5 groups of 16 prototypes
    const float* pbase[NWAY];
#pragma unroll
    for (int m = 0; m < NWAY; ++m)
        pbase[m] = prm + (size_t)(tile * PTILE + m * 16 + row) * DIM + khalf;

    v8f acc[NWAY] = {};
    float qn = 0.0f;
    float pn[NWAY] = {0.0f, 0.0f, 0.0f, 0.0f, 0.0f};

    // K loop: each step consumes K=4 (f32 WMMA 16x16x4); norms ride on the same loads.
#pragma unroll 4
    for (int kk = 0; kk < DIM; kk += 4) {
        v2f a = *(const v2f*)(qbase + kk);                 // global_load_b64
        qn = fmaf(a.x, a.x, qn);
        qn = fmaf(a.y, a.y, qn);
#pragma unroll
        for (int m = 0; m < NWAY; ++m) {
            v2f b = *(const v2f*)(pbase[m] + kk);          // global_load_b64
            pn[m] = fmaf(b.x, b.x, pn[m]);
            pn[m] = fmaf(b.y, b.y, pn[m]);
            // v_wmma_f32_16x16x4_f32 : exact fp32 MAC, D = A x B + C
            acc[m] = __builtin_amdgcn_wmma_f32_16x16x4_f32(
                false, a, false, b, (short)0, acc[m], false, false);
        }
    }

    // Combine K-halves of the norms (lane L <-> lane L^16 hold complementary halves)
    qn += __shfl_xor(qn, 16, 32);
#pragma unroll
    for (int m = 0; m < NWAY; ++m)
        pn[m] += __shfl_xor(pn[m], 16, 32);

    __shared__ float Cs[NWAY][16][16];
    __shared__ float qns[QTILE];
    __shared__ float pns[PTILE];

    if (lane < 16) {
        qns[row] = qn;
#pragma unroll
        for (int m = 0; m < NWAY; ++m) pns[m * 16 + row] = pn[m];
    }
    // C/D layout (ISA 7.12.2): VGPR r holds M=r (lanes 0-15) and M=8+r (lanes 16-31), N=lane&15
    const int mbase = (lane >> 4) << 3;
#pragma unroll
    for (int m = 0; m < NWAY; ++m)
#pragma unroll
        for (int r = 0; r < 8; ++r)
            Cs[m][mbase + r][row] = acc[m][r];
    __syncthreads();

    float lossv = 0.0f, corr = 0.0f;
    if (lane < 16) {
        const int i  = lane;
        const int gq = tile * QTILE + i;
        const float qn2 = qns[i];
        float s[NWAY];
#pragma unroll
        for (int k = 0; k < NWAY; ++k) {
            const int j   = i * NWAY + k;
            const float dot  = Cs[j >> 4][i][j & 15];
            const float dist = qn2 + pns[j] - 2.0f * dot;
            s[k] = -dist;                       // logits = -dist
        }
        float mx = s[0];
#pragma unroll
        for (int k = 1; k < NWAY; ++k) mx = fmaxf(mx, s[k]);
        float sum = 0.0f;
#pragma unroll
        for (int k = 0; k < NWAY; ++k) sum += __expf(s[k] - mx);
        const float lse = mx + __logf(sum);

        const float* lr = lab + (size_t)gq * NWAY;
        int   yhat = 0, ytrue = 0;
        float best = s[0], bl = lr[0], dotlab = (s[0] - lse) * lr[0];
#pragma unroll
        for (int k = 1; k < NWAY; ++k) {
            const float l = lr[k];
            dotlab += (s[k] - lse) * l;
            if (s[k] > best) { best = s[k]; yhat = k; }   // first-max tiebreak, like argmax
            if (l    > bl)   { bl   = l;    ytrue = k; }
        }
        lossv = -dotlab;
        corr  = (yhat == ytrue) ? 1.0f : 0.0f;
    }

    // Deterministic wave butterfly reduction over all 32 lanes (upper lanes carry zeros)
#pragma unroll
    for (int off = 16; off >= 1; off >>= 1) {
        lossv += __shfl_xor(lossv, off, 32);
        corr  += __shfl_xor(corr,  off, 32);
    }
    if (lane == 0) {
        partial[2 * tile + 0] = lossv;   // full overwrite each call -> no ws zero-init needed
        partial[2 * tile + 1] = corr;
    }
}

// Single-wave, fixed-order final reduction -> deterministic output
__global__ __launch_bounds__(32) void proto_finalize_kernel(
    const float* __restrict__ partial, int nblocks, float inv_n,
    float* __restrict__ out)
{
    const int lane = threadIdx.x;
    float l = 0.0f, c = 0.0f;
    for (int b = lane; b < nblocks; b += 32) {
        l += partial[2 * b + 0];
        c += partial[2 * b + 1];
    }
#pragma unroll
    for (int off = 16; off >= 1; off >>= 1) {
        l += __shfl_xor(l, off, 32);
        c += __shfl_xor(c, off, 32);
    }
    if (lane == 0) {
        out[0] = l * inv_n;   // loss
        out[1] = c * inv_n;   // accuracy
    }
}

extern "C" void kernel_launch(void* const* d_in, const int* in_sizes, int n_in,
                              void* d_out, int out_size, void* d_ws, size_t ws_size,
                              hipStream_t stream) {
    const float* qry = (const float*)d_in[0];   // [n, 256] fp32
    const float* prm = (const float*)d_in[1];   // [n*5, 256] fp32
    const float* lab = (const float*)d_in[2];   // [n, 5] fp32 one-hot
    float* out = (float*)d_out;                 // [2] fp32 (loss, acc)

    const int n     = in_sizes[2] / NWAY;       // 8192
    const int tiles = n / QTILE;                // 512

    float* partial = (float*)d_ws;              // tiles*2 floats, fully overwritten

    proto_main_kernel<<<tiles, 32, 0, stream>>>(qry, prm, lab, partial);
    proto_finalize_kernel<<<1, 32, 0, stream>>>(partial, tiles, 1.0f / (float)n, out);
}